// FCOSDecoder_17317308137873
// MI455X (gfx1250) — compile-verified
//
#include <hip/hip_runtime.h>
#include <hip/hip_bf16.h>

#define B_ 16
#define C_ 80
#define K_ 1000
#define L_ 21824
#define NBIN 4096
#define CANDMAX 2048
#define NT 1024

typedef float v2f __attribute__((ext_vector_type(2)));
typedef float v8f __attribute__((ext_vector_type(8)));
typedef unsigned int u32;
typedef unsigned long long u64;
typedef u32 u32x4 __attribute__((ext_vector_type(4)));
typedef int  i32x4 __attribute__((ext_vector_type(4)));
typedef int  i32x8 __attribute__((ext_vector_type(8)));

// workspace layout (bytes)
#define OFF_HIST    0u          // B*4096*4 = 262144
#define OFF_SEL     262144u     // 1024
#define OFF_CANDCNT 263168u     // 1024
#define OFF_CAND    264192u     // B*2048*8 = 262144
#define OFF_X1      526336u
#define OFF_Y1      591872u
#define OFF_X2      657408u
#define OFF_Y2      722944u
#define OFF_AREA    788480u
#define OFF_SCORE   854016u
#define OFF_CLS     919552u
#define OFF_VALID   985088u
#define OFF_SUP     1050624u    // B*1024*32*4 = 2097152

__device__ __forceinline__ u32 okey(float x){
  u32 u = __float_as_uint(x);
  return u ^ ((u32)(((int)u) >> 31) | 0x80000000u);
}
__device__ __forceinline__ float inv_okey(u32 k){
  u32 u = (k & 0x80000000u) ? (k ^ 0x80000000u) : ~k;
  return __uint_as_float(u);
}
__device__ __forceinline__ float sigm(float x){ return 1.0f/(1.0f + __expf(-x)); }

__device__ __forceinline__ u32 ballot32(bool p){
#if __has_builtin(__builtin_amdgcn_ballot_w32)
  return __builtin_amdgcn_ballot_w32(p);
#else
  return (u32)__ballot((int)p);
#endif
}

// ---------- Pass A/B: streaming decode + radix histogram (chunked per block) ----------
// Each block owns a contiguous chunk of ONE batch so the LDS histogram aggregates
// thousands of samples before the (tiny) global flush.
__global__ void fcos_hist_kernel(const float* __restrict__ cls, const float* __restrict__ ctr,
                                 int mask /*hw2-1*/, int npb /*C_*hw2*/, int chunk,
                                 u32* __restrict__ hist, const u32* __restrict__ sel, int pass){
  __shared__ u32 sh[NBIN];
  int tid = threadIdx.x;
  for (int j = tid; j < NBIN; j += 256) sh[j] = 0;
  __syncthreads();
  int nbpb = npb / chunk;                 // blocks per batch
  int b    = blockIdx.x / nbpb;
  int base = b * npb + (blockIdx.x % nbpb) * chunk;
  u32 sel1 = (pass == 1) ? sel[b*8 + 0] : 0u;
  const float* ctrb = ctr + b * (mask + 1);
  for (int off = tid; off < chunk; off += 256){
    int i = base + off;
    __builtin_prefetch(cls + i + 16384, 0, 0);     // global_prefetch_b8 ahead of the stream
    float p      = sigm(cls[i]);
    float cen    = sigm(ctrb[i & mask]);
    float masked = (p >= 0.05f) ? __builtin_sqrtf(p*cen) : -1.0f;
    u32 k = okey(masked);
    if (pass == 0)                 atomicAdd(&sh[k >> 20], 1u);
    else if ((k >> 20) == sel1)    atomicAdd(&sh[(k >> 8) & 0xFFFu], 1u);
  }
  __syncthreads();
  for (int j = tid; j < NBIN; j += 256){
    u32 v = sh[j];
    if (v) atomicAdd(&hist[b*NBIN + j], v);
  }
}

// ---------- Threshold scan: parallel staging, short serial walk ----------
__global__ void fcos_scan_kernel(const u32* __restrict__ hist, u32* __restrict__ sel, int pass){
  __shared__ u32 part[256];
  __shared__ u32 lsh[NBIN];
  int b = blockIdx.x, tid = threadIdx.x;
  u32 s = 0;
#pragma unroll
  for (int j = 0; j < 16; ++j){
    u32 v = hist[b*NBIN + tid*16 + j];
    lsh[tid*16 + j] = v;
    s += v;
  }
  part[tid] = s;
  __syncthreads();
  if (tid == 0){
    u32 need = (pass == 0) ? (u32)K_ : ((u32)K_ - sel[b*8 + 1]);
    u32 acc = 0; int seg = 0;
    for (int t = 255; t >= 0; --t){
      if (acc + part[t] >= need){ seg = t; break; }
      acc += part[t];
    }
    u32 found = (u32)(seg*16);
    for (int bin = seg*16 + 15; bin >= seg*16; --bin){
      u32 c = lsh[bin];
      if (acc + c >= need){ found = (u32)bin; break; }
      acc += c;
    }
    if (pass == 0){ sel[b*8 + 0] = found; sel[b*8 + 1] = acc; }
    else          { sel[b*8 + 2] = (sel[b*8 + 0] << 12) | found; }
  }
}

// ---------- Pass C: compact candidates >= 24-bit threshold ----------
__global__ void fcos_compact_kernel(const float* __restrict__ cls, const float* __restrict__ ctr,
                                    int mask, int sh2, int loff, const u32* __restrict__ sel,
                                    u32* __restrict__ cnt, u64* __restrict__ cand){
  int i = blockIdx.x * 256 + threadIdx.x;
  int pos = i & mask;
  int t   = i >> sh2;
  int c   = t % C_;
  int b   = t / C_;
  float p      = sigm(cls[i]);
  float cen    = sigm(ctr[b*(mask+1) + pos]);
  float masked = (p >= 0.05f) ? __builtin_sqrtf(p*cen) : -1.0f;
  u32 k = okey(masked);
  if ((k >> 8) >= sel[b*8 + 2]){
    u32 slot = atomicAdd(&cnt[b], 1u);
    if (slot < CANDMAX){
      u32 idx = (u32)(c * L_ + loff + pos);
      cand[b*CANDMAX + slot] = ((u64)k << 32) | (u32)(~idx);  // key desc, idx asc
    }
  }
}

// ---------- Sort candidates (bitonic in LDS), gather boxes for top-K ----------
__global__ void fcos_sortgather_kernel(const u64* __restrict__ cand, const u32* __restrict__ cnt,
    const float* __restrict__ rg0, const float* __restrict__ rg1, const float* __restrict__ rg2,
    const float* __restrict__ rg3, const float* __restrict__ rg4,
    float* __restrict__ X1, float* __restrict__ Y1, float* __restrict__ X2, float* __restrict__ Y2,
    float* __restrict__ AREA, float* __restrict__ SCORE, u32* __restrict__ CLS, u32* __restrict__ VALID){
  __shared__ u64 sk[CANDMAX];
  int b = blockIdx.x, tid = threadIdx.x;
  u32 cn = cnt[b];
  int n = (cn < (u32)CANDMAX) ? (int)cn : CANDMAX;
  for (int t = tid; t < CANDMAX; t += 1024) sk[t] = (t < n) ? cand[b*CANDMAX + t] : 0ULL;
  __syncthreads();
  for (int k = 2; k <= CANDMAX; k <<= 1){
    for (int j = k >> 1; j > 0; j >>= 1){
      for (int t = tid; t < CANDMAX; t += 1024){
        int ixj = t ^ j;
        if (ixj > t){
          u64 a = sk[t], c = sk[ixj];
          bool desc = ((t & k) == 0);
          if (desc ? (a < c) : (a > c)){ sk[t] = c; sk[ixj] = a; }
        }
      }
      __syncthreads();
    }
  }
  int j = tid;
  if (j < NT){
    u64 e  = sk[j];
    u32 k32 = (u32)(e >> 32);
    bool ok = (j < K_) && ((k32 & 0x80000000u) != 0);  // value >= 0 and within K
    float score = -1.f, x1 = 0.f, y1 = 0.f, x2 = 0.f, y2 = 0.f, area = 0.f;
    u32 cid = 0x40000000u + (u32)j;                    // unique class for pads
    if (ok){
      u32 idx = ~((u32)e);
      score = inv_okey(k32);
      cid = idx / (u32)L_;
      int loc = (int)(idx % (u32)L_);
      const int offs[5] = {0, 16384, 20480, 21504, 21760};
      const int szs[5]  = {128, 64, 32, 16, 8};
      const int strd[5] = {8, 16, 32, 64, 128};
      int lvl = 4;
      if      (loc < offs[1]) lvl = 0;
      else if (loc < offs[2]) lvl = 1;
      else if (loc < offs[3]) lvl = 2;
      else if (loc < offs[4]) lvl = 3;
      int pos = loc - offs[lvl];
      int hw = szs[lvl], hw2 = hw*hw;
      int yy = pos / hw, xx = pos % hw;
      float s  = (float)strd[lvl];
      float cx = xx*s + 0.5f*s, cy = yy*s + 0.5f*s;
      const float* rg = (lvl == 0) ? rg0 : (lvl == 1) ? rg1 : (lvl == 2) ? rg2 : (lvl == 3) ? rg3 : rg4;
      float lv = rg[(b*4 + 0)*hw2 + pos];
      float tv = rg[(b*4 + 1)*hw2 + pos];
      float rv = rg[(b*4 + 2)*hw2 + pos];
      float bv = rg[(b*4 + 3)*hw2 + pos];
      x1 = cx - lv; y1 = cy - tv; x2 = cx + rv; y2 = cy + bv;
      area = fmaxf(x2 - x1, 0.f) * fmaxf(y2 - y1, 0.f);
    }
    int o = b*NT + j;
    X1[o] = x1; Y1[o] = y1; X2[o] = x2; Y2[o] = y2;
    AREA[o] = area; SCORE[o] = score; CLS[o] = cid; VALID[o] = ok ? 1u : 0u;
  }
}

// ---------- Suppression bitmask: WMMA computes pairwise area sums ----------
// Each wave: rows [i0, i0+16) x cols [j0, j0+32). D = A(16x4) x B(4x16):
//   A[m][0]=area_row[m], A[m][1]=1   ->  D[m][n] = area_row[m] + area_col[n]
__global__ void fcos_sup_kernel(const float* __restrict__ X1, const float* __restrict__ Y1,
                                const float* __restrict__ X2, const float* __restrict__ Y2,
                                const float* __restrict__ AREA, const u32* __restrict__ CLS,
                                u32* __restrict__ SUP){
  int tid  = threadIdx.x;
  int gid  = blockIdx.x * 256 + tid;
  int wave = gid >> 5;
  int lane = tid & 31;
  int b  = wave >> 11;          // 2048 waves per batch
  int t  = wave & 2047;
  int i0 = (t >> 5) << 4;       // 64 row-tiles of 16
  int wj = t & 31;              // 32 col-words of 32
  int j0 = wj << 5;
  bool lo = lane < 16;

  int cA = b*NT + j0 + (lane & 15);   // column for D0
  int cB = cA + 16;                   // column for D1
  float ax1 = X1[cA], ay1 = Y1[cA], ax2 = X2[cA], ay2 = Y2[cA]; u32 ca = CLS[cA];
  float bx1 = X1[cB], by1 = Y1[cB], bx2 = X2[cB], by2 = Y2[cB]; u32 cb = CLS[cB];

  v2f av, b0, b1;
  v8f cz = {0.f,0.f,0.f,0.f,0.f,0.f,0.f,0.f};
  av.x = lo ? AREA[b*NT + i0 + lane] : 0.f;  av.y = lo ? 1.f : 0.f;   // K0,K1 (lanes 0-15); K2,K3 = 0
  b0.x = lo ? 1.f : 0.f;                     b0.y = lo ? AREA[cA] : 0.f;
  b1.x = lo ? 1.f : 0.f;                     b1.y = lo ? AREA[cB] : 0.f;

  v8f D0 = __builtin_amdgcn_wmma_f32_16x16x4_f32(false, av, false, b0, (short)0, cz, false, false);
  v8f D1 = __builtin_amdgcn_wmma_f32_16x16x4_f32(false, av, false, b1, (short)0, cz, false, false);

#pragma unroll
  for (int r = 0; r < 8; ++r){
    int ri = i0 + r + (lo ? 0 : 8);
    int ro = b*NT + ri;
    float rx1 = X1[ro], ry1 = Y1[ro], rx2 = X2[ro], ry2 = Y2[ro]; u32 cr = CLS[ro];

    float iw0 = fminf(rx2, ax2) - fmaxf(rx1, ax1);
    float ih0 = fminf(ry2, ay2) - fmaxf(ry1, ay1);
    float in0 = fmaxf(iw0, 0.f) * fmaxf(ih0, 0.f);
    float un0 = fmaxf(D0[r] - in0, 1e-9f);                // union = areasum - inter
    bool p0 = (in0 > 0.5f*un0) && (cr == ca) && ((j0 + (lane & 15)) > ri);

    float iw1 = fminf(rx2, bx2) - fmaxf(rx1, bx1);
    float ih1 = fminf(ry2, by2) - fmaxf(ry1, by1);
    float in1 = fmaxf(iw1, 0.f) * fmaxf(ih1, 0.f);
    float un1 = fmaxf(D1[r] - in1, 1e-9f);
    bool p1 = (in1 > 0.5f*un1) && (cr == cb) && ((j0 + 16 + (lane & 15)) > ri);

    u32 m0 = ballot32(p0);   // low16: row r cols j0..15 ; high16: row r+8 cols j0..15
    u32 m1 = ballot32(p1);   // same for cols j0+16..31
    if (lane == 0)  SUP[(b*NT + i0 + r    )*32 + wj] = (m0 & 0xFFFFu) | (m1 << 16);
    if (lane == 16) SUP[(b*NT + i0 + r + 8)*32 + wj] = (m0 >> 16)     | (m1 & 0xFFFF0000u);
  }
}

// ---------- NMS scan (TDM loads 128KB sup matrix into LDS) + output ----------
__global__ void fcos_nms_kernel(const u32* __restrict__ SUP, const u32* __restrict__ VALID,
    const float* __restrict__ X1, const float* __restrict__ Y1,
    const float* __restrict__ X2, const float* __restrict__ Y2,
    const float* __restrict__ SCORE, const u32* __restrict__ CLS, float* __restrict__ out){
  extern __shared__ u32 sup_s[];        // NT*32 u32 = 128 KB (dynamic LDS)
  __shared__ u32 keepw[32];
  int b = blockIdx.x, tid = threadIdx.x;
  const u32* gsup = SUP + (size_t)b * NT * 32;

#if __has_builtin(__builtin_amdgcn_tensor_load_to_lds)
  if (tid < 32){
    // Tensor DMA: 1D tile of 32768 dwords, global -> LDS
    u32 lds_base = (u32)(u64)(void*)sup_s;          // generic LDS addr low bits = LDS offset
    u64 ga = (u64)gsup;
    u32x4 g0 = { 1u,                                 // count=1
                 lds_base,
                 (u32)ga,
                 (((u32)(ga >> 32)) & 0x01FFFFFFu) | 0x80000000u };  // addr[56:32] | type=2
    i32x8 g1 = { (int)0x00020000,                    // data_size=2 (4B)
                 (int)0x80000000u,                   // tensor_dim0 = 32768 (bits 63:48)
                 (int)0x00010000,                    // tensor_dim1 = 1 (bits 95:80)
                 (int)0x80000000u,                   // tile_dim0 = 32768 (bits 127:112)
                 0,                                  // tile_dim1/2 = 0 (unused)
                 32768,                              // tensor_dim0_stride (bits 191:160)
                 0, 0 };
    i32x4 z4 = {0,0,0,0};
#if __clang_major__ >= 23
    i32x8 z8 = {0,0,0,0,0,0,0,0};
    __builtin_amdgcn_tensor_load_to_lds(g0, g1, z4, z4, z8, 0);
#else
    __builtin_amdgcn_tensor_load_to_lds(g0, g1, z4, z4, 0);
#endif
#if __has_builtin(__builtin_amdgcn_s_wait_tensorcnt)
    __builtin_amdgcn_s_wait_tensorcnt(0);
#else
    asm volatile("s_wait_tensorcnt 0x0" ::: "memory");
#endif
  }
  __syncthreads();
#else
  for (int t = tid; t < NT*32; t += 1024) sup_s[t] = gsup[t];
  __syncthreads();
#endif

  // Sequential keep-scan on one wave: lane owns one 32-bit keep word.
  if (tid < 32){
    u32 kw = 0;
#pragma unroll
    for (int bit = 0; bit < 32; ++bit)
      kw |= (VALID[b*NT + tid*32 + bit] & 1u) << bit;
    for (int i = 0; i < K_; ++i){
      u32 kword = (u32)__shfl((int)kw, i >> 5, 32);  // broadcast keep word holding bit i
      if ((kword >> (i & 31)) & 1u)
        kw &= ~sup_s[i*32 + tid];
    }
    keepw[tid] = kw;
  }
  __syncthreads();

  if (tid < K_){
    bool kp = (keepw[tid >> 5] >> (tid & 31)) & 1u;
    int o = b*NT + tid;
    float* op = out + ((size_t)b * K_ + tid) * 6;
    if (kp){
      op[0] = X1[o]; op[1] = Y1[o]; op[2] = X2[o]; op[3] = Y2[o];
      op[4] = SCORE[o]; op[5] = (float)CLS[o];
    } else {
      op[0] = -1.f; op[1] = -1.f; op[2] = -1.f; op[3] = -1.f; op[4] = -1.f; op[5] = -1.f;
    }
  }
}

// ---------- Host ----------
extern "C" void kernel_launch(void* const* d_in, const int* in_sizes, int n_in,
                              void* d_out, int out_size, void* d_ws, size_t ws_size,
                              hipStream_t stream){
  const float* cls[5]; const float* reg[5]; const float* ctr[5];
  for (int l = 0; l < 5; ++l){
    cls[l] = (const float*)d_in[3*l + 0];
    reg[l] = (const float*)d_in[3*l + 1];
    ctr[l] = (const float*)d_in[3*l + 2];
  }
  char* ws = (char*)d_ws;
  u32* hist  = (u32*)(ws + OFF_HIST);
  u32* sel   = (u32*)(ws + OFF_SEL);
  u32* cnt   = (u32*)(ws + OFF_CANDCNT);
  u64* cand  = (u64*)(ws + OFF_CAND);
  float* X1  = (float*)(ws + OFF_X1);
  float* Y1  = (float*)(ws + OFF_Y1);
  float* X2  = (float*)(ws + OFF_X2);
  float* Y2  = (float*)(ws + OFF_Y2);
  float* AREA= (float*)(ws + OFF_AREA);
  float* SC  = (float*)(ws + OFF_SCORE);
  u32* CLS   = (u32*)(ws + OFF_CLS);
  u32* VAL   = (u32*)(ws + OFF_VALID);
  u32* SUP   = (u32*)(ws + OFF_SUP);

  const int hw2s[5]  = {16384, 4096, 1024, 256, 64};
  const int sh2s[5]  = {14, 12, 10, 8, 6};
  const int loffs[5] = {0, 16384, 20480, 21504, 21760};
  const int nbpb[5]  = {40, 10, 4, 1, 1};          // hist blocks per batch
  // chunk = C_*hw2 / nbpb : {32768, 32768, 20480, 20480, 5120} — all multiples of 256

  hipMemsetAsync(hist, 0, (size_t)B_*NBIN*4, stream);
  hipMemsetAsync(ws + OFF_SEL, 0, 2048, stream);   // sel + candcnt

  for (int l = 0; l < 5; ++l){
    int npb = C_*hw2s[l];
    fcos_hist_kernel<<<B_*nbpb[l], 256, 0, stream>>>(cls[l], ctr[l], hw2s[l]-1, npb,
                                                     npb/nbpb[l], hist, sel, 0);
  }
  fcos_scan_kernel<<<B_, 256, 0, stream>>>(hist, sel, 0);

  hipMemsetAsync(hist, 0, (size_t)B_*NBIN*4, stream);
  for (int l = 0; l < 5; ++l){
    int npb = C_*hw2s[l];
    fcos_hist_kernel<<<B_*nbpb[l], 256, 0, stream>>>(cls[l], ctr[l], hw2s[l]-1, npb,
                                                     npb/nbpb[l], hist, sel, 1);
  }
  fcos_scan_kernel<<<B_, 256, 0, stream>>>(hist, sel, 1);

  for (int l = 0; l < 5; ++l){
    int blocks = (B_*C_*hw2s[l]) / 256;
    fcos_compact_kernel<<<blocks, 256, 0, stream>>>(cls[l], ctr[l], hw2s[l]-1, sh2s[l],
                                                    loffs[l], sel, cnt, cand);
  }

  fcos_sortgather_kernel<<<B_, 1024, 0, stream>>>(cand, cnt,
      reg[0], reg[1], reg[2], reg[3], reg[4],
      X1, Y1, X2, Y2, AREA, SC, CLS, VAL);

  fcos_sup_kernel<<<(B_*2048*32)/256, 256, 0, stream>>>(X1, Y1, X2, Y2, AREA, CLS, SUP);

  fcos_nms_kernel<<<B_, 1024, NT*32*sizeof(u32), stream>>>(SUP, VAL, X1, Y1, X2, Y2, SC, CLS,
                                                           (float*)d_out);
}